// BaselineEdgePredictor_52819507806805
// MI455X (gfx1250) — compile-verified
//
#include <hip/hip_runtime.h>

#define N_NODES   1000000
#define E_EDGES   262144
#define NFC       9
#define CARD      50000
#define EMB       128
#define EDGE_DIM  27
#define TILE_E    16

typedef __attribute__((ext_vector_type(2))) float v2f;
typedef __attribute__((ext_vector_type(8))) float v8f;

// Sum the active embedding rows for one node over 8 consecutive emb columns.
// Active feature columns by type: t==0 -> h 0..3, t==1 -> h 4, t==2 -> h 5..8
// (equivalent to the reference's mask-multiply, skipping the zeroed terms).
__device__ __forceinline__ void encode8(const int* __restrict__ x,
                                        const float* __restrict__ emb_type,
                                        const float* __restrict__ emb_feats,
                                        int node, int n0, float acc[8]) {
    const int* row = x + (size_t)node * 10;
    const int t = row[0];
    const float* et = emb_type + (size_t)t * EMB + n0;
    float4 a0 = *(const float4*)(et);
    float4 a1 = *(const float4*)(et + 4);

    int hb, he;
    if (t == 0)      { hb = 0; he = 4; }
    else if (t == 1) { hb = 4; he = 5; }
    else             { hb = 5; he = 9; }

    for (int h = hb; h < he; ++h) {
        const int idx = row[1 + h];
        const float* ef = emb_feats + ((size_t)h * CARD + (size_t)idx) * EMB + n0;
        float4 b0 = *(const float4*)(ef);
        float4 b1 = *(const float4*)(ef + 4);
        a0.x += b0.x; a0.y += b0.y; a0.z += b0.z; a0.w += b0.w;
        a1.x += b1.x; a1.y += b1.y; a1.z += b1.z; a1.w += b1.w;
    }
    acc[0] = a0.x; acc[1] = a0.y; acc[2] = a0.z; acc[3] = a0.w;
    acc[4] = a1.x; acc[5] = a1.y; acc[6] = a1.z; acc[7] = a1.w;
}

__global__ __launch_bounds__(256)
void edge_pred_fused(const int* __restrict__ x,
                     const int* __restrict__ src,
                     const int* __restrict__ dst,
                     const int* __restrict__ neg_dst,
                     const float* __restrict__ msg,
                     const float* __restrict__ emb_type,
                     const float* __restrict__ emb_feats,
                     const float* __restrict__ edge_w,
                     const float* __restrict__ edge_b,
                     const float* __restrict__ out_w,
                     const float* __restrict__ out_b,
                     float* __restrict__ out) {
    __shared__ float tmsg[TILE_E * EMB];   // 8 KB: staged target_msg tile

    const int tid  = threadIdx.x;
    const int wave = tid >> 5;             // 0..7 -> emb column tile
    const int lane = tid & 31;
    const int e0   = blockIdx.x * TILE_E;

    // ---------------- Phase 1: target_msg tile via V_WMMA_F32_16X16X4_F32 ----
    {
        const int m  = lane & 15;          // A-matrix row (edge within tile)
        const int hi = lane >> 4;          // K sub-phase select (ISA A/B layout)
        const int n  = wave * 16 + (lane & 15); // output emb column 0..127
        const float* arow = msg    + (size_t)(e0 + m) * EDGE_DIM; // A row
        const float* brow = edge_w + (size_t)n        * EDGE_DIM; // B[k][n]=edge_w[n][k]
        v8f c = {};

        // 6 full K-steps: k in {0,4,8,12,16,20}; max index = 20+2*1+1 = 23 < 27
        #pragma unroll
        for (int k = 0; k < 24; k += 4) {
            const int kk = k + 2 * hi;
            v2f a, b;
            a.x = arow[kk];  a.y = arow[kk + 1];
            b.x = brow[kk];  b.y = brow[kk + 1];
            c = __builtin_amdgcn_wmma_f32_16x16x4_f32(
                    false, a, false, b, (short)0, c, false, false);
        }
        // Tail K-step (k=24): kk in {24,26} valid; kk+1 in {25,27}, 27 is pad.
        {
            const int kk = 24 + 2 * hi;
            const bool ok = (kk + 1 < EDGE_DIM);
            const int  k1 = ok ? (kk + 1) : (EDGE_DIM - 1);  // clamped, in-bounds
            v2f a, b;
            a.x = arow[kk];
            b.x = brow[kk];
            const float ay = arow[k1];                       // unconditional load
            const float by = brow[k1];                       // unconditional load
            a.y = ok ? ay : 0.0f;                            // v_cndmask, no branch
            b.y = ok ? by : 0.0f;
            c = __builtin_amdgcn_wmma_f32_16x16x4_f32(
                    false, a, false, b, (short)0, c, false, false);
        }

        const float bias = edge_b[n];
        #pragma unroll
        for (int r = 0; r < 8; ++r) {      // C/D layout: VGPR r -> M = r + 8*hi
            const int mm = r + 8 * hi;
            tmsg[mm * EMB + n] = c[r] + bias;
        }
    }
    __syncthreads();

    // ---------------- Phase 2: gathers + ReLU + dot with out_w ---------------
    const int el  = tid >> 4;              // local edge 0..15
    const int c16 = tid & 15;              // 16 threads share one edge
    const int n0  = c16 * 8;               // 8 consecutive emb columns
    const int eg  = e0 + el;

    const int ns = src[eg];
    const int np = dst[eg];
    const int nn = neg_dst[eg];

    float hs[8], hp[8], hn[8];
    encode8(x, emb_type, emb_feats, ns, n0, hs);
    encode8(x, emb_type, emb_feats, np, n0, hp);
    encode8(x, emb_type, emb_feats, nn, n0, hn);

    const float4 tm0 = *(const float4*)&tmsg[el * EMB + n0];
    const float4 tm1 = *(const float4*)&tmsg[el * EMB + n0 + 4];
    const float tmv[8] = { tm0.x, tm0.y, tm0.z, tm0.w, tm1.x, tm1.y, tm1.z, tm1.w };

    const float4 w0 = *(const float4*)(out_w + n0);
    const float4 w1 = *(const float4*)(out_w + n0 + 4);
    const float wv[8] = { w0.x, w0.y, w0.z, w0.w, w1.x, w1.y, w1.z, w1.w };

    float pacc = 0.0f, nacc = 0.0f;
    #pragma unroll
    for (int j = 0; j < 8; ++j) {
        float p = hs[j] + hp[j]; p = p > 0.0f ? p : 0.0f;   // relu
        float q = hs[j] + hn[j]; q = q > 0.0f ? q : 0.0f;   // relu
        pacc = fmaf(p + tmv[j], wv[j], pacc);
        nacc = fmaf(q + tmv[j], wv[j], nacc);
    }

    // Reduce across the 16 threads of this edge (stays inside wave32 halves).
    #pragma unroll
    for (int off = 8; off >= 1; off >>= 1) {
        pacc += __shfl_xor(pacc, off, 32);
        nacc += __shfl_xor(nacc, off, 32);
    }
    if (c16 == 0) {
        const float b = out_b[0];
        out[eg]           = pacc + b;   // pos
        out[E_EDGES + eg] = nacc + b;   // neg
    }
}

extern "C" void kernel_launch(void* const* d_in, const int* in_sizes, int n_in,
                              void* d_out, int out_size, void* d_ws, size_t ws_size,
                              hipStream_t stream) {
    const int*   x         = (const int*)  d_in[0];
    const int*   src       = (const int*)  d_in[1];
    const int*   dst       = (const int*)  d_in[2];
    const int*   neg_dst   = (const int*)  d_in[3];
    const float* msg       = (const float*)d_in[4];
    const float* emb_type  = (const float*)d_in[5];
    const float* emb_feats = (const float*)d_in[6];
    const float* edge_w    = (const float*)d_in[7];
    const float* edge_b    = (const float*)d_in[8];
    const float* out_w     = (const float*)d_in[9];
    const float* out_b     = (const float*)d_in[10];
    float*       out       = (float*)      d_out;

    dim3 grid(E_EDGES / TILE_E);   // 16384 blocks
    dim3 block(256);               // 8 wave32 per block
    edge_pred_fused<<<grid, block, 0, stream>>>(x, src, dst, neg_dst, msg,
                                                emb_type, emb_feats,
                                                edge_w, edge_b, out_w, out_b, out);
}